// WarpingLayer_51153060496180
// MI455X (gfx1250) — compile-verified
//
#include <hip/hip_runtime.h>
#include <hip/hip_bf16.h>

// Problem dims from the reference setup_inputs(): x:[B,C,H,W], flow:[B,2,H,W]
constexpr int Bn  = 8;
constexpr int Cn  = 64;
constexpr int Hn  = 256;   // power of two -> shift/mask index math
constexpr int Wn  = 512;   // power of two
constexpr int HWn = Hn * Wn;

// One thread per (b,h,w) pixel; loops over all C channels reusing the
// per-pixel bilinear weights and clamped tap offsets.
__global__ __launch_bounds__(256, 1) void warp_bilinear_kernel(
    const float* __restrict__ x,
    const float* __restrict__ flow,
    float* __restrict__ out)
{
    const int tid = blockIdx.x * blockDim.x + threadIdx.x;   // 0 .. B*H*W-1
    const int w = tid & (Wn - 1);
    const int h = (tid >> 9) & (Hn - 1);                     // Wn = 2^9
    const int b = tid >> 17;                                 // Hn*Wn = 2^17

    const size_t pix = (size_t)h * Wn + w;

    // flow is read exactly once -> non-temporal loads (TH=NT), don't cache it.
    const float* fp  = flow + (size_t)b * 2 * HWn + pix;
    const float  fxp = __builtin_nontemporal_load(fp);
    const float  fyp = __builtin_nontemporal_load(fp + HWn);

    // Reference math, step for step (align_corners=False unnormalize).
    const float fx = fxp * (2.0f / (Wn - 1.0f));
    const float fy = fyp * (2.0f / (Hn - 1.0f));
    const float bx = -1.0f + (float)w * (2.0f / (Wn - 1.0f));
    const float by = -1.0f + (float)h * (2.0f / (Hn - 1.0f));
    const float gx = bx + fx;
    const float gy = by + fy;
    const float ixf = ((gx + 1.0f) * (float)Wn - 1.0f) * 0.5f;
    const float iyf = ((gy + 1.0f) * (float)Hn - 1.0f) * 0.5f;

    const float ix0f = floorf(ixf);
    const float iy0f = floorf(iyf);
    const float ix1f = ix0f + 1.0f;
    const float iy1f = iy0f + 1.0f;
    const float wx1 = ixf - ix0f, wx0 = 1.0f - wx1;
    const float wy1 = iyf - iy0f, wy0 = 1.0f - wy1;

    // Validity on UNCLAMPED coords (matches reference), weights zeroed if OOB.
    const bool vx0 = (ix0f >= 0.0f) && (ix0f < (float)Wn);
    const bool vx1 = (ix1f >= 0.0f) && (ix1f < (float)Wn);
    const bool vy0 = (iy0f >= 0.0f) && (iy0f < (float)Hn);
    const bool vy1 = (iy1f >= 0.0f) && (iy1f < (float)Hn);

    const float w00 = (vx0 && vy0) ? wx0 * wy0 : 0.0f;
    const float w01 = (vx1 && vy0) ? wx1 * wy0 : 0.0f;
    const float w10 = (vx0 && vy1) ? wx0 * wy1 : 0.0f;
    const float w11 = (vx1 && vy1) ? wx1 * wy1 : 0.0f;

    // Clamped integer tap coordinates (always in-bounds loads).
    const int x0 = (int)fminf(fmaxf(ix0f, 0.0f), (float)(Wn - 1));
    const int x1 = (int)fminf(fmaxf(ix1f, 0.0f), (float)(Wn - 1));
    const int y0 = (int)fminf(fmaxf(iy0f, 0.0f), (float)(Hn - 1));
    const int y1 = (int)fminf(fmaxf(iy1f, 0.0f), (float)(Hn - 1));

    const int o00 = y0 * Wn + x0;
    const int o01 = y0 * Wn + x1;
    const int o10 = y1 * Wn + x0;
    const int o11 = y1 * Wn + x1;

    const float* __restrict__ xb = x   + (size_t)b * Cn * HWn;
    float*       __restrict__ ob = out + (size_t)b * Cn * HWn + pix;

    // Stream all 64 channels with the hoisted weights/offsets. Unroll 8:
    // 32 outstanding gathers per batch per lane for latency hiding while
    // staying well under the 6-bit LOADcnt ceiling.
#pragma unroll 8
    for (int c = 0; c < Cn; ++c) {
        const float* __restrict__ xc = xb + (size_t)c * HWn;
        float acc = xc[o00] * w00;
        acc = fmaf(xc[o01], w01, acc);
        acc = fmaf(xc[o10], w10, acc);
        acc = fmaf(xc[o11], w11, acc);
        // Output is write-once -> NT store: keep the 192MB L2 for x's
        // gather reuse instead of polluting it with the 256MB output stream.
        __builtin_nontemporal_store(acc, ob + (size_t)c * HWn);
    }
}

extern "C" void kernel_launch(void* const* d_in, const int* in_sizes, int n_in,
                              void* d_out, int out_size, void* d_ws, size_t ws_size,
                              hipStream_t stream) {
    const float* x    = (const float*)d_in[0];   // [8,64,256,512] f32
    const float* flow = (const float*)d_in[1];   // [8,2,256,512]  f32
    float*       out  = (float*)d_out;           // [8,64,256,512] f32

    constexpr int npix    = Bn * Hn * Wn;        // 1,048,576 pixels
    constexpr int threads = 256;                 // 8 waves (wave32)
    warp_bilinear_kernel<<<npix / threads, threads, 0, stream>>>(x, flow, out);
}